// OnlineClustering_18064632447419
// MI455X (gfx1250) — compile-verified
//
#include <hip/hip_runtime.h>
#include <math.h>

typedef float v2f __attribute__((ext_vector_type(2)));
typedef float v8f __attribute__((ext_vector_type(8)));

#define IN_DIM   768
#define OUT_DIM  4096
#define N_TOK    16384      // 16 * 1024
#define TOK_PB   1024
#define N_BATCH  16
#define TGT_INV_T (1.0f/0.06f)
#define PRED_INV_T (1.0f/0.12f)
#define EXP_MAXV 50.0f
#define SK_EPS   1e-8f
#define NORM_EPS 1e-7f

// workspace layout (bytes)
#define INV_OFF    ((size_t)0)                               // 16384 * 4      = 64 KB
#define LOGITS_OFF ((size_t)65536)                           // 16384*4096*4   = 256 MB
#define COL_OFF    (LOGITS_OFF + (size_t)N_TOK*OUT_DIM*4)    // 16*4096*4      = 256 KB
#define LOSS_OFF   (COL_OFF + (size_t)N_BATCH*OUT_DIM*4)     // 4 B

// ---------------- row inverse norms: one wave per row ----------------
__global__ __launch_bounds__(256) void k_invnorm(const float* __restrict__ x,
                                                 float* __restrict__ invn) {
    int wv = threadIdx.x >> 5, lane = threadIdx.x & 31;
    int row = blockIdx.x * 8 + wv;
    size_t base = (size_t)row * IN_DIM;
    float ss = 0.f;
#pragma unroll
    for (int i = 0; i < 24; ++i) {
        float v = x[base + lane + i * 32];
        ss = fmaf(v, v, ss);
    }
#pragma unroll
    for (int off = 16; off >= 1; off >>= 1) ss += __shfl_xor(ss, off, 32);
    if (lane == 0) invn[row] = 1.f / fmaxf(sqrtf(ss), NORM_EPS);
}

// ---------------- WMMA f32 GEMM: logits = diag(invn) * X * W^T + bias ----------------
// block: 256 thr = 8 waves; block tile 128(M) x 64(N); K chunks of 32 staged in LDS,
// software-pipelined: next chunk's global loads issued before current chunk's WMMAs.
__global__ __launch_bounds__(256) void k_gemm(const float* __restrict__ x,
                                              const float* __restrict__ w,
                                              const float* __restrict__ bias,
                                              const float* __restrict__ invn,
                                              float* __restrict__ logits) {
    __shared__ __align__(16) float sA[128 * 34];
    __shared__ __align__(16) float sB[64 * 34];
    const int t    = threadIdx.x;
    const int lane = t & 31, wv = t >> 5;
    const int half = lane >> 4, l16 = lane & 15;
    const int m0 = blockIdx.y * 128, n0 = blockIdx.x * 64;

    v8f acc[4] = {};
    float ra[16], rb[8];

    auto load_chunk = [&](int kc) {
#pragma unroll
        for (int i = 0; i < 16; ++i) {
            int idx = t + i * 256;
            ra[i] = x[(size_t)(m0 + (idx >> 5)) * IN_DIM + kc + (idx & 31)];
        }
#pragma unroll
        for (int i = 0; i < 8; ++i) {
            int idx = t + i * 256;
            rb[i] = w[(size_t)(n0 + (idx >> 5)) * IN_DIM + kc + (idx & 31)];
        }
    };

    load_chunk(0);
    for (int kc = 0; kc < IN_DIM; kc += 32) {
#pragma unroll
        for (int i = 0; i < 16; ++i) {
            int idx = t + i * 256;
            sA[(idx >> 5) * 34 + (idx & 31)] = ra[i];
        }
#pragma unroll
        for (int i = 0; i < 8; ++i) {
            int idx = t + i * 256;
            sB[(idx >> 5) * 34 + (idx & 31)] = rb[i];
        }
        __syncthreads();
        if (kc + 32 < IN_DIM) load_chunk(kc + 32);   // overlap with WMMAs below
        // A frag (16x4 f32): lanes 0-15 K={kk,kk+1}, lanes 16-31 K={kk+2,kk+3}
#pragma unroll
        for (int kk = 0; kk < 32; kk += 4) {
            v2f a = *(const v2f*)&sA[(wv * 16 + l16) * 34 + kk + half * 2];
#pragma unroll
            for (int j = 0; j < 4; ++j) {
                v2f b = *(const v2f*)&sB[(j * 16 + l16) * 34 + kk + half * 2];
                acc[j] = __builtin_amdgcn_wmma_f32_16x16x4_f32(
                    false, a, false, b, (short)0, acc[j], false, false);
            }
        }
        __syncthreads();
    }
    // fold in row norms + bias, store (C layout: VGPR r -> M = half*8 + r, lane N = l16)
    float inr[8];
#pragma unroll
    for (int r = 0; r < 8; ++r) inr[r] = invn[m0 + wv * 16 + half * 8 + r];
#pragma unroll
    for (int j = 0; j < 4; ++j) {
        int n = n0 + j * 16 + l16;
        float bv = bias[n];
#pragma unroll
        for (int r = 0; r < 8; ++r) {
            int m = m0 + wv * 16 + half * 8 + r;
            logits[(size_t)m * OUT_DIM + n] = acc[j][r] * inr[r] + bv;
        }
    }
}

// ---------------- column max over tokens (per batch), float4, scaled to target temp ----------------
__global__ __launch_bounds__(256) void k_colmax(const float* __restrict__ logits,
                                                float* __restrict__ col) {
    int b = blockIdx.x >> 2;                              // 4 blocks per batch
    int k = ((blockIdx.x & 3) << 10) + threadIdx.x * 4;   // 1024 cols per block
    const float* p = logits + (size_t)b * TOK_PB * OUT_DIM + k;
    float4 m = make_float4(-3.402823466e38f, -3.402823466e38f,
                           -3.402823466e38f, -3.402823466e38f);
    for (int n = 0; n < TOK_PB; ++n) {
        float4 v = *(const float4*)(p + (size_t)n * OUT_DIM);
        m.x = fmaxf(m.x, v.x); m.y = fmaxf(m.y, v.y);
        m.z = fmaxf(m.z, v.z); m.w = fmaxf(m.w, v.w);
    }
    m.x *= TGT_INV_T; m.y *= TGT_INV_T; m.z *= TGT_INV_T; m.w *= TGT_INV_T;
    *(float4*)(col + b * OUT_DIM + k) = m;                // shift = max(logits/T)
}

// ---------------- M = exp(logits/T + 50 - shift), vectorized float4 ----------------
__global__ __launch_bounds__(256) void k_exp(const float* __restrict__ logits,
                                             const float* __restrict__ col,
                                             float* __restrict__ M) {
    size_t base = ((size_t)blockIdx.x * 256 + threadIdx.x) * 4;
    int k = (int)(base & (OUT_DIM - 1));
    int b = (int)(base >> 22);               // 1024*4096 = 2^22
    float4 lg = *(const float4*)(logits + base);
    float4 c4 = *(const float4*)(col + b * OUT_DIM + k);
    float4 o;
    o.x = expf(lg.x * TGT_INV_T + EXP_MAXV - c4.x);
    o.y = expf(lg.y * TGT_INV_T + EXP_MAXV - c4.y);
    o.z = expf(lg.z * TGT_INV_T + EXP_MAXV - c4.z);
    o.w = expf(lg.w * TGT_INV_T + EXP_MAXV - c4.w);
    *(float4*)(M + base) = o;
}

// ---------------- column sums over tokens (per batch), float4 ----------------
__global__ __launch_bounds__(256) void k_colsum(const float* __restrict__ M,
                                                float* __restrict__ col) {
    int b = blockIdx.x >> 2;
    int k = ((blockIdx.x & 3) << 10) + threadIdx.x * 4;
    const float* p = M + (size_t)b * TOK_PB * OUT_DIM + k;
    float4 s = make_float4(0.f, 0.f, 0.f, 0.f);
    for (int n = 0; n < TOK_PB; ++n) {
        float4 v = *(const float4*)(p + (size_t)n * OUT_DIM);
        s.x += v.x; s.y += v.y; s.z += v.z; s.w += v.w;
    }
    *(float4*)(col + b * OUT_DIM + k) = s;
}

__device__ __forceinline__ float block_sum(float v, float* red) {
#pragma unroll
    for (int off = 16; off >= 1; off >>= 1) v += __shfl_xor(v, off, 32);
    int lane = threadIdx.x & 31, wv = threadIdx.x >> 5;
    if (lane == 0) red[wv] = v;
    __syncthreads();
    float tot = (threadIdx.x < 8) ? red[threadIdx.x] : 0.f;
    tot += __shfl_xor(tot, 4, 32);
    tot += __shfl_xor(tot, 2, 32);
    tot += __shfl_xor(tot, 1, 32);
    if (threadIdx.x == 0) red[0] = tot;
    __syncthreads();
    float r = red[0];
    __syncthreads();
    return r;
}

__device__ __forceinline__ float block_max(float v, float* red) {
#pragma unroll
    for (int off = 16; off >= 1; off >>= 1) v = fmaxf(v, __shfl_xor(v, off, 32));
    int lane = threadIdx.x & 31, wv = threadIdx.x >> 5;
    if (lane == 0) red[wv] = v;
    __syncthreads();
    float tot = (threadIdx.x < 8) ? red[threadIdx.x] : -3.402823466e38f;
    tot = fmaxf(tot, __shfl_xor(tot, 4, 32));
    tot = fmaxf(tot, __shfl_xor(tot, 2, 32));
    tot = fmaxf(tot, __shfl_xor(tot, 1, 32));
    if (threadIdx.x == 0) red[0] = tot;
    __syncthreads();
    float r = red[0];
    __syncthreads();
    return r;
}

// ---------------- fused column-divide + row-normalize (one block per row, float4) ----------------
__global__ __launch_bounds__(256) void k_rownorm(float* __restrict__ M,
                                                 const float* __restrict__ col) {
    __shared__ float red[8];
    int row = blockIdx.x;
    const float4* cb = (const float4*)(col + (row >> 10) * OUT_DIM);
    float4* rp = (float4*)(M + (size_t)row * OUT_DIM);
    float4 v[4];
    float s = 0.f;
#pragma unroll
    for (int i = 0; i < 4; ++i) {
        int c = threadIdx.x + i * 256;     // float4 index, 1024 per row
        float4 m4 = rp[c];
        float4 c4 = cb[c];
        m4.x /= (c4.x + SK_EPS); m4.y /= (c4.y + SK_EPS);
        m4.z /= (c4.z + SK_EPS); m4.w /= (c4.w + SK_EPS);
        v[i] = m4;
        s += m4.x + m4.y + m4.z + m4.w;
    }
    float rowsum = block_sum(s, red);
    float inv = 1.f / (rowsum + SK_EPS);
#pragma unroll
    for (int i = 0; i < 4; ++i) {
        int c = threadIdx.x + i * 256;
        float4 o = v[i];
        o.x *= inv; o.y *= inv; o.z *= inv; o.w *= inv;
        rp[c] = o;
    }
}

// ---------------- loss: -mean_n sum_k tgt * log_softmax(logits/T2) ----------------
__global__ void k_zero(float* p) { if (threadIdx.x == 0) *p = 0.f; }

__global__ __launch_bounds__(256) void k_loss_row(const float* __restrict__ logits,
                                                  const float* __restrict__ tgt,
                                                  float* __restrict__ acc) {
    __shared__ float red[8];
    int row = blockIdx.x;
    const float4* lp = (const float4*)(logits + (size_t)row * OUT_DIM);
    const float4* tp = (const float4*)(tgt + (size_t)row * OUT_DIM);
    float4 s[4];
    float mx = -3.402823466e38f;
#pragma unroll
    for (int i = 0; i < 4; ++i) {
        float4 l4 = lp[threadIdx.x + i * 256];
        l4.x *= PRED_INV_T; l4.y *= PRED_INV_T; l4.z *= PRED_INV_T; l4.w *= PRED_INV_T;
        s[i] = l4;
        mx = fmaxf(mx, fmaxf(fmaxf(l4.x, l4.y), fmaxf(l4.z, l4.w)));
    }
    float rmax = block_max(mx, red);
    float se = 0.f;
#pragma unroll
    for (int i = 0; i < 4; ++i)
        se += expf(s[i].x - rmax) + expf(s[i].y - rmax) +
              expf(s[i].z - rmax) + expf(s[i].w - rmax);
    float tot = block_sum(se, red);
    float lse = logf(tot) + rmax;
    float a = 0.f;
#pragma unroll
    for (int i = 0; i < 4; ++i) {
        float4 t4 = tp[threadIdx.x + i * 256];
        a = fmaf(t4.x, s[i].x - lse, a);
        a = fmaf(t4.y, s[i].y - lse, a);
        a = fmaf(t4.z, s[i].z - lse, a);
        a = fmaf(t4.w, s[i].w - lse, a);
    }
    float rowloss = block_sum(a, red);
    if (threadIdx.x == 0) atomicAdd(acc, rowloss);
}

__global__ void k_loss_final(const float* __restrict__ acc, float* __restrict__ out) {
    if (threadIdx.x == 0 && blockIdx.x == 0)
        out[0] = -(*acc) * (1.0f / (float)N_TOK);   // PREFACTOR = 1
}

extern "C" void kernel_launch(void* const* d_in, const int* in_sizes, int n_in,
                              void* d_out, int out_size, void* d_ws, size_t ws_size,
                              hipStream_t stream) {
    const float* x    = (const float*)d_in[0];
    const float* w    = (const float*)d_in[1];
    const float* bias = (const float*)d_in[2];
    char* ws = (char*)d_ws;
    float* invn    = (float*)(ws + INV_OFF);
    float* logits  = (float*)(ws + LOGITS_OFF);
    float* col     = (float*)(ws + COL_OFF);
    float* lossacc = (float*)(ws + LOSS_OFF);
    float* assign  = (float*)d_out;                         // [16,1024,4096] f32
    float* lossout = assign + (size_t)N_TOK * OUT_DIM;      // scalar loss

    k_invnorm<<<N_TOK / 8, 256, 0, stream>>>(x, invn);
    dim3 gg(OUT_DIM / 64, N_TOK / 128);
    k_gemm<<<gg, 256, 0, stream>>>(x, w, bias, invn, logits);
    k_colmax<<<N_BATCH * 4, 256, 0, stream>>>(logits, col);
    k_exp<<<(int)(((size_t)N_TOK * OUT_DIM) / 1024), 256, 0, stream>>>(logits, col, assign);
    for (int it = 0; it < 3; ++it) {
        k_colsum<<<N_BATCH * 4, 256, 0, stream>>>(assign, col);
        k_rownorm<<<N_TOK, 256, 0, stream>>>(assign, col);
    }
    k_zero<<<1, 1, 0, stream>>>(lossacc);
    k_loss_row<<<N_TOK, 256, 0, stream>>>(logits, assign, lossacc);
    k_loss_final<<<1, 1, 0, stream>>>(lossacc, lossout);
}